// GPDconv_41188736368644
// MI455X (gfx1250) — compile-verified
//
#include <hip/hip_runtime.h>
#include <math.h>

#define BSZ 8
#define NN 65536
#define NUM_PTS 4096
#define KK 32
#define EDGES (BSZ * NUM_PTS * KK) // 1048576

typedef __attribute__((ext_vector_type(2))) float v2f;
typedef __attribute__((ext_vector_type(8))) float v8f;

// ---------------- zero fill ----------------
__global__ void zero_f32(float* __restrict__ p, int n) {
    int i = blockIdx.x * blockDim.x + threadIdx.x;
    int stride = gridDim.x * blockDim.x;
    for (; i < n; i += stride) p[i] = 0.0f;
}

// ---------------- (b,32,N) -> (b,N,32) ----------------
__global__ void transpose_c_to_n(const float* __restrict__ in, float* __restrict__ out) {
    __shared__ float tile[32][33];
    int b = blockIdx.y;
    int n0 = blockIdx.x * 32;
    int tx = threadIdx.x, ty = threadIdx.y; // (32,8)
#pragma unroll
    for (int i = 0; i < 4; ++i) {
        int c = ty + 8 * i;
        tile[c][tx] = in[(size_t)b * 32 * NN + (size_t)c * NN + n0 + tx];
    }
    __syncthreads();
#pragma unroll
    for (int i = 0; i < 4; ++i) {
        int n = ty + 8 * i;
        out[((size_t)b * NN + n0 + n) * 32 + tx] = tile[tx][n];
    }
}

// ---------------- (b,N,32) -> (b,32,N) ----------------
__global__ void transpose_n_to_c(const float* __restrict__ in, float* __restrict__ out) {
    __shared__ float tile[32][33];
    int b = blockIdx.y;
    int n0 = blockIdx.x * 32;
    int tx = threadIdx.x, ty = threadIdx.y;
#pragma unroll
    for (int i = 0; i < 4; ++i) {
        int n = ty + 8 * i;
        tile[n][tx] = in[((size_t)b * NN + n0 + n) * 32 + tx];
    }
    __syncthreads();
#pragma unroll
    for (int i = 0; i < 4; ++i) {
        int c = ty + 8 * i;
        out[(size_t)b * 32 * NN + (size_t)c * NN + n0 + tx] = tile[tx][c];
    }
}

// ---------------- gauss + k-normalized wg; one wave per (b,p), lane=k ----------------
__global__ void gauss_kernel(const float* __restrict__ grid, const float* __restrict__ gweight,
                             const int* __restrict__ eg, const int* __restrict__ ega,
                             const float* __restrict__ basepts, const float* __restrict__ bw,
                             float* __restrict__ gauss, float* __restrict__ wgn) {
    int wid = (blockIdx.x * blockDim.x + threadIdx.x) >> 5; // b*4096 + p
    int lane = threadIdx.x & 31;                            // k
    int b = wid >> 12;
    int p = wid & 4095;
    int e = wid * 32 + lane; // flat (b,p,k) edge id
    int n = eg[e];
    int q = ega[e];
    float g0 = grid[((size_t)b * NN + n) * 2 + 0];
    float g1 = grid[((size_t)b * NN + n) * 2 + 1];
    float d0 = g0 - basepts[q * 2 + 0];
    float d1 = g1 - basepts[q * 2 + 1];
    float dw = bw[p * 2 + 0] * d0 * d0 + bw[p * 2 + 1] * d1 * d1;
    float g = expf(-dw);
    float wg = g * gweight[(size_t)b * NN + n];
    float s = wg * wg;
#pragma unroll
    for (int off = 16; off > 0; off >>= 1) s += __shfl_xor(s, off, 32);
    float norm = sqrtf(s) + 1e-5f;
    gauss[e] = g;
    wgn[e] = wg / norm;
}

// ---------------- value2features scatter: wave per edge, lane=channel ----------------
__global__ void scatter_v2f(const float* __restrict__ xT, const float* __restrict__ wgn,
                            const int* __restrict__ eg, const int* __restrict__ ega,
                            float* __restrict__ xh) {
    int w = (blockIdx.x * blockDim.x + threadIdx.x) >> 5; // flat (b,p,k)
    int c = threadIdx.x & 31;
    int b = w >> 17; // / (4096*32)
    int n = eg[w];
    int q = ega[w];
    float v = wgn[w] * xT[((size_t)b * NN + n) * 32 + c];
    atomicAdd(&xh[((size_t)b * NUM_PTS + q) * 32 + c], v);
}

// ---------------- M[q,(i,o)] = sum_j weights[i,o,j]*D[j,q] via f32 WMMA ----------------
__global__ void wmma_M(const float* __restrict__ D, const float* __restrict__ W,
                       float* __restrict__ M) {
    int wid = (blockIdx.x * blockDim.x + threadIdx.x) >> 5;
    int lane = threadIdx.x & 31;
    int half = lane >> 4;
    int ll = lane & 15;
    int qt = wid >> 6, iot = wid & 63;
    int q0 = qt * 16, io0 = iot * 16;
    v8f acc = {};
#pragma unroll
    for (int jb = 0; jb < 16; jb += 4) {
        int ja = jb + (half ? 2 : 0);
        v2f a, bb;
        // A = D^T tile: rows q, cols j
        a.x = D[(size_t)ja * NUM_PTS + q0 + ll];
        a.y = D[(size_t)(ja + 1) * NUM_PTS + q0 + ll];
        // B: rows j, cols io ; weights flat (i,o,j) => W[io*16 + j]
        int io = io0 + ll;
        bb.x = W[io * 16 + ja];
        bb.y = W[io * 16 + ja + 1];
        acc = __builtin_amdgcn_wmma_f32_16x16x4_f32(false, a, false, bb, (short)0, acc,
                                                    false, false);
    }
#pragma unroll
    for (int r = 0; r < 8; ++r) {
        M[(size_t)(q0 + r + (half ? 8 : 0)) * 1024 + io0 + ll] = acc[r];
    }
}

// ---------------- per-point Y_q(8x32) = X_q(8x32) * M_q(32x32), batch padded to 16 ----------------
__global__ void wmma_apply(const float* __restrict__ xh, const float* __restrict__ M,
                           float* __restrict__ yv) {
    int q = blockIdx.x * (blockDim.x >> 5) + (threadIdx.x >> 5);
    int lane = threadIdx.x & 31;
    int half = lane >> 4;
    int ll = lane & 15;
    const float* Mq = M + (size_t)q * 1024;
    v8f acc0 = {}, acc1 = {};
#pragma unroll
    for (int ib = 0; ib < 32; ib += 4) {
        int ia = ib + (half ? 2 : 0);
        v2f a, b0, b1;
        float a0 = 0.0f, a1 = 0.0f;
        if (ll < 8) { // A rows 8..15 are zero padding
            a0 = xh[((size_t)ll * NUM_PTS + q) * 32 + ia];
            a1 = xh[((size_t)ll * NUM_PTS + q) * 32 + ia + 1];
        }
        a.x = a0; a.y = a1;
        b0.x = Mq[ia * 32 + ll];       b0.y = Mq[(ia + 1) * 32 + ll];
        b1.x = Mq[ia * 32 + 16 + ll];  b1.y = Mq[(ia + 1) * 32 + 16 + ll];
        acc0 = __builtin_amdgcn_wmma_f32_16x16x4_f32(false, a, false, b0, (short)0, acc0,
                                                     false, false);
        acc1 = __builtin_amdgcn_wmma_f32_16x16x4_f32(false, a, false, b1, (short)0, acc1,
                                                     false, false);
    }
    if (half == 0) { // valid rows 0..7 live in lanes 0-15
#pragma unroll
        for (int r = 0; r < 8; ++r) {
            yv[((size_t)r * NUM_PTS + q) * 32 + ll] = acc0[r];
            yv[((size_t)r * NUM_PTS + q) * 32 + 16 + ll] = acc1[r];
        }
    }
}

// ---------------- features2value scatter ----------------
__global__ void scatter_f2v(const float* __restrict__ yv, const float* __restrict__ gauss,
                            const int* __restrict__ eg, const int* __restrict__ ega,
                            float* __restrict__ outT) {
    int w = (blockIdx.x * blockDim.x + threadIdx.x) >> 5;
    int c = threadIdx.x & 31;
    int b = w >> 17;
    int n = eg[w];
    int q = ega[w];
    float v = gauss[w] * yv[((size_t)b * NUM_PTS + q) * 32 + c];
    atomicAdd(&outT[((size_t)b * NN + n) * 32 + c], v);
}

extern "C" void kernel_launch(void* const* d_in, const int* in_sizes, int n_in,
                              void* d_out, int out_size, void* d_ws, size_t ws_size,
                              hipStream_t stream) {
    const float* x        = (const float*)d_in[0];
    const float* grid     = (const float*)d_in[1];
    const float* gweight  = (const float*)d_in[2];
    const int*   eg       = (const int*)d_in[3];
    const int*   ega      = (const int*)d_in[4];
    const float* basepts  = (const float*)d_in[5];
    const float* bw       = (const float*)d_in[6];
    const float* D        = (const float*)d_in[7];
    const float* W        = (const float*)d_in[8];
    float* out = (float*)d_out;

    char* ws = (char*)d_ws;
    size_t off = 0;
    auto alloc = [&](size_t bytes) {
        char* p = ws + off;
        off += (bytes + 255) & ~(size_t)255;
        return p;
    };
    float* big   = (float*)alloc((size_t)BSZ * NN * 32 * 4);      // xT, later reused as outT
    float* gauss = (float*)alloc((size_t)EDGES * 4);
    float* wgn   = (float*)alloc((size_t)EDGES * 4);
    float* xh    = (float*)alloc((size_t)BSZ * NUM_PTS * 32 * 4);
    float* Mbuf  = (float*)alloc((size_t)NUM_PTS * 1024 * 4);
    float* yv    = (float*)alloc((size_t)BSZ * NUM_PTS * 32 * 4);

    // 1) x (b,32,N) -> xT (b,N,32)
    transpose_c_to_n<<<dim3(NN / 32, BSZ), dim3(32, 8), 0, stream>>>(x, big);
    // 2) edge gaussians + k-normalized weights
    gauss_kernel<<<(BSZ * NUM_PTS) / 8, 256, 0, stream>>>(grid, gweight, eg, ega, basepts, bw,
                                                          gauss, wgn);
    // 3) x_hat scatter
    zero_f32<<<1024, 256, 0, stream>>>(xh, BSZ * NUM_PTS * 32);
    scatter_v2f<<<EDGES / 8, 256, 0, stream>>>(big, wgn, eg, ega, xh);
    // 4) M = weights x D (WMMA f32)
    wmma_M<<<(256 * 64) / 8, 256, 0, stream>>>(D, W, Mbuf);
    // 5) y = x_hat x M_q per point (WMMA f32)
    wmma_apply<<<NUM_PTS / 8, 256, 0, stream>>>(xh, Mbuf, yv);
    // 6) output scatter into zeroed (b,N,32) accumulator (reuses xT space)
    zero_f32<<<4096, 256, 0, stream>>>(big, BSZ * NN * 32);
    scatter_f2v<<<EDGES / 8, 256, 0, stream>>>(yv, gauss, eg, ega, big);
    // 7) (b,N,32) -> (b,32,N)
    transpose_n_to_c<<<dim3(NN / 32, BSZ), dim3(32, 8), 0, stream>>>(big, out);
}